// DirectedHyperConvNetwork_26070451486833
// MI455X (gfx1250) — compile-verified
//
#include <hip/hip_runtime.h>
#include <stdint.h>
#include <stddef.h>

#define N_POIS   100000
#define N_HYPER  50000
#define NNZ_TOT  1600000
#define DIM      256

typedef __attribute__((ext_vector_type(2))) float v2f;
typedef __attribute__((ext_vector_type(4))) float v4f;
typedef __attribute__((ext_vector_type(8))) float v8f;

// ---------------------------------------------------------------------------
// Async global->LDS copy of one dword (CDNA5 GLOBAL_LOAD_ASYNC_TO_LDS_B32,
// GVS mode: 64-bit SGPR base + 32-bit VGPR byte offset). Tracked by ASYNCcnt.
// ---------------------------------------------------------------------------
__device__ __forceinline__ void async_cp_dword(void* lds_dst, const void* gbase,
                                               int byte_off) {
  unsigned l = (unsigned)(size_t)lds_dst;   // low 32 bits == LDS byte address
  asm volatile("global_load_async_to_lds_b32 %0, %1, %2"
               :: "v"(l), "v"(byte_off), "s"(gbase) : "memory");
}
__device__ __forceinline__ void wait_async_le1() {
  asm volatile("s_wait_asynccnt 1" ::: "memory");
}
__device__ __forceinline__ void wait_async_0() {
  asm volatile("s_wait_asynccnt 0" ::: "memory");
}

// ---------------------------------------------------------------------------
// SpMM scatter: dst[rows[e], :] += vals[e] * src[cols[e], :]
// COO streams staged into LDS with double-buffered async copies; 64 threads
// per nonzero (one float4 of the 256-wide row each); f32 hardware atomics.
// Dense operands are L2-resident (192 MB L2), so gathers/atomics hit L2.
// ---------------------------------------------------------------------------
#define CHUNK 64
#define TPB   256

__global__ __launch_bounds__(TPB) void spmm_scatter_kernel(
    const int* __restrict__ rows, const int* __restrict__ cols,
    const float* __restrict__ vals, const float* __restrict__ src,
    float* __restrict__ dst, int nnz, int nchunks)
{
  __shared__ int   s_rows[2][CHUNK];
  __shared__ int   s_cols[2][CHUNK];
  __shared__ float s_vals[2][CHUNK];

  const int t = threadIdx.x;

  // Stage one chunk (waves 0-1: rows, 2-3: cols, 4-5: vals; full-EXEC waves).
  auto issue_stage = [&](int chunk, int buf) {
    int base = chunk * CHUNK;
    if (t < CHUNK) {
      int g = base + t; if (g >= nnz) g = nnz - 1;
      async_cp_dword(&s_rows[buf][t], rows, g * 4);
    } else if (t < 2 * CHUNK) {
      int i = t - CHUNK;
      int g = base + i; if (g >= nnz) g = nnz - 1;
      async_cp_dword(&s_cols[buf][i], cols, g * 4);
    } else if (t < 3 * CHUNK) {
      int i = t - 2 * CHUNK;
      int g = base + i; if (g >= nnz) g = nnz - 1;
      async_cp_dword(&s_vals[buf][i], vals, g * 4);
    }
  };

  int c = blockIdx.x;
  if (c >= nchunks) return;

  int buf = 0;
  issue_stage(c, 0);

  const int ch = t & 63;   // which float4 of the 256-wide row
  const int qb = t >> 6;   // 4 nonzeros processed concurrently per block

  for (; c < nchunks; c += gridDim.x) {
    int nextc = c + gridDim.x;
    if (nextc < nchunks) { issue_stage(nextc, buf ^ 1); wait_async_le1(); }
    else                 { wait_async_0(); }
    __syncthreads();

    const int base = c * CHUNK;
    #pragma unroll 4
    for (int step = 0; step < CHUNK / 4; ++step) {
      int local = step * 4 + qb;
      int g = base + local;
      if (g < nnz) {
        int   r  = s_rows[buf][local];
        int   cc = s_cols[buf][local];
        float v  = s_vals[buf][local];
        const v4f* sp = (const v4f*)(src + (size_t)cc * DIM) + ch;
        v4f x = *sp;                                  // coalesced b128, L2 hit
        float* dp = dst + (size_t)r * DIM + ch * 4;
        unsafeAtomicAdd(dp + 0, v * x.x);             // global_atomic_add_f32
        unsafeAtomicAdd(dp + 1, v * x.y);
        unsafeAtomicAdd(dp + 2, v * x.z);
        unsafeAtomicAdd(dp + 3, v * x.w);
      }
    }
    __syncthreads();
    buf ^= 1;
  }
}

// ---------------------------------------------------------------------------
// Fused layer update (in place): m = msg_src buffer (scatter result)
//   embs = (relu(m) * drop1 + prev) * drop2
// Drop masks are streamed once -> non-temporal loads keep them out of L2,
// preserving residency of the embedding/message buffers.
// ---------------------------------------------------------------------------
__global__ __launch_bounds__(256) void fuse_layer_kernel(
    const float* __restrict__ prev, const float* __restrict__ d1,
    const float* __restrict__ d2, float* __restrict__ embs, int total4)
{
  int i = blockIdx.x * 256 + threadIdx.x;
  if (i >= total4) return;
  v4f m = ((const v4f*)embs)[i];
  v4f p = ((const v4f*)prev)[i];
  v4f a = __builtin_nontemporal_load(((const v4f*)d1) + i);
  v4f b = __builtin_nontemporal_load(((const v4f*)d2) + i);
  v4f o;
  o.x = (fmaxf(m.x, 0.f) * a.x + p.x) * b.x;
  o.y = (fmaxf(m.y, 0.f) * a.y + p.y) * b.y;
  o.z = (fmaxf(m.z, 0.f) * a.z + p.z) * b.z;
  o.w = (fmaxf(m.w, 0.f) * a.w + p.w) * b.w;
  ((v4f*)embs)[i] = o;
}

// ---------------------------------------------------------------------------
// Final combine on the matrix pipe:
//   out = w0*E0 + w1*E1 + w2*E2   (w = softmax(layer_attention), 3 terms)
// Per 16x16 fp32 tile (one wave): D = sum_l  A(=E_l tile) x (w_l * I)
// via chained V_WMMA_F32_16X16X4_F32. e2_out is read then overwritten
// (each tile owned by exactly one wave -> no hazard).
// A 16x4 f32 layout: lanes0-15 M=lane, v0=K0 v1=K1; lanes16-31 v0=K2 v1=K3.
// B 4x16  f32 layout: lanes0-15 N=lane, v0=K0 v1=K1 rows; lanes16-31 K2/K3.
// C/D 16x16: lane -> N, vgpr r -> M = r + 8*(lane>=16).
// ---------------------------------------------------------------------------
__global__ __launch_bounds__(256) void combine_wmma_kernel(
    const float* __restrict__ e0, const float* __restrict__ e1,
    float* __restrict__ e2_out, const float* __restrict__ att)
{
  // 3-way softmax computed per-thread (3 scalar loads, negligible)
  float a0 = att[0], a1 = att[1], a2 = att[2];
  float mx = fmaxf(a0, fmaxf(a1, a2));
  float x0 = __expf(a0 - mx), x1 = __expf(a1 - mx), x2 = __expf(a2 - mx);
  float inv = 1.0f / (x0 + x1 + x2);
  float w[3] = {x0 * inv, x1 * inv, x2 * inv};

  const int lane = threadIdx.x & 31;
  const int wib  = threadIdx.x >> 5;
  const int tile = blockIdx.x * 8 + wib;          // 100000 tiles exactly
  const int tiles_per_row = DIM / 16;             // 16
  const int m0 = (tile / tiles_per_row) * 16;
  const int d0 = (tile % tiles_per_row) * 16;

  const int mrow = lane & 15;                     // M for A, N for B/D
  const int kh   = lane >> 4;

  const float* srcs[3] = {e0, e1, e2_out};
  v8f acc = {};
  #pragma unroll
  for (int l = 0; l < 3; ++l) {
    const float* S = srcs[l] + (size_t)(m0 + mrow) * DIM + d0;
    #pragma unroll
    for (int j = 0; j < 4; ++j) {
      const int k0 = 4 * j + 2 * kh;              // global K of this lane's pair
      v2f a, b;
      a.x = S[k0];
      a.y = S[k0 + 1];
      b.x = (mrow == k0)     ? w[l] : 0.0f;       // (w_l * I) slice, rows k0,k0+1
      b.y = (mrow == k0 + 1) ? w[l] : 0.0f;
      acc = __builtin_amdgcn_wmma_f32_16x16x4_f32(
          /*neg_a=*/false, a, /*neg_b=*/false, b,
          /*c_mod=*/(short)0, acc, /*reuse_a=*/false, /*reuse_b=*/false);
    }
  }

  float* O = e2_out + (size_t)m0 * DIM + d0 + mrow;
  #pragma unroll
  for (int r = 0; r < 8; ++r)
    O[(size_t)(r + 8 * kh) * DIM] = acc[r];
}

// ---------------------------------------------------------------------------
extern "C" void kernel_launch(void* const* d_in, const int* in_sizes, int n_in,
                              void* d_out, int out_size, void* d_ws, size_t ws_size,
                              hipStream_t stream) {
  (void)in_sizes; (void)n_in; (void)out_size; (void)ws_size;

  const float* pois      = (const float*)d_in[0];
  const float* tar_vals  = (const float*)d_in[1];
  const float* src_vals  = (const float*)d_in[2];
  const float* att       = (const float*)d_in[3];
  const float* drop1     = (const float*)d_in[4];
  const float* drop2     = (const float*)d_in[5];
  const int*   tar_rows  = (const int*)d_in[6];
  const int*   tar_cols  = (const int*)d_in[7];
  const int*   src_rows  = (const int*)d_in[8];
  const int*   src_cols  = (const int*)d_in[9];

  const size_t ND = (size_t)N_POIS * DIM;
  const size_t HD = (size_t)N_HYPER * DIM;

  float* msg_tar = (float*)d_ws;                  // H*D  (51.2 MB)
  float* embs1   = (float*)d_ws + HD;             // N*D  (102.4 MB)
  float* outp    = (float*)d_out;                 // N*D; doubles as layer-2 acc

  const int nchunks   = (NNZ_TOT + CHUNK - 1) / CHUNK;   // 25000
  const int spmm_grid = 6250;
  const int total4    = (int)(ND / 4);                   // 6,400,000
  const int elem_grid = (total4 + 255) / 256;
  const int comb_grid = (int)(ND / 256 / 8);             // 12500 (8 tiles/block)

  // ---- layer 0 ----
  (void)hipMemsetAsync(msg_tar, 0, HD * sizeof(float), stream);
  spmm_scatter_kernel<<<spmm_grid, TPB, 0, stream>>>(
      tar_rows, tar_cols, tar_vals, pois, msg_tar, NNZ_TOT, nchunks);
  (void)hipMemsetAsync(embs1, 0, ND * sizeof(float), stream);
  spmm_scatter_kernel<<<spmm_grid, TPB, 0, stream>>>(
      src_rows, src_cols, src_vals, msg_tar, embs1, NNZ_TOT, nchunks);
  fuse_layer_kernel<<<elem_grid, 256, 0, stream>>>(
      pois, drop1, drop2, embs1, total4);

  // ---- layer 1 ----
  (void)hipMemsetAsync(msg_tar, 0, HD * sizeof(float), stream);
  spmm_scatter_kernel<<<spmm_grid, TPB, 0, stream>>>(
      tar_rows, tar_cols, tar_vals, embs1, msg_tar, NNZ_TOT, nchunks);
  (void)hipMemsetAsync(outp, 0, ND * sizeof(float), stream);
  spmm_scatter_kernel<<<spmm_grid, TPB, 0, stream>>>(
      src_rows, src_cols, src_vals, msg_tar, outp, NNZ_TOT, nchunks);
  fuse_layer_kernel<<<elem_grid, 256, 0, stream>>>(
      embs1, drop1 + ND, drop2 + ND, outp, total4);

  // ---- softmax-weighted combine on the matrix pipe ----
  combine_wmma_kernel<<<comb_grid, 256, 0, stream>>>(pois, embs1, outp, att);
}